// CorrelationLoss_55035710931775
// MI455X (gfx1250) — compile-verified
//
#include <hip/hip_runtime.h>

typedef float v2f  __attribute__((ext_vector_type(2)));
typedef float v8f  __attribute__((ext_vector_type(8)));
typedef unsigned int u32x4 __attribute__((ext_vector_type(4)));
typedef int   i32x4 __attribute__((ext_vector_type(4)));
typedef int   i32x8 __attribute__((ext_vector_type(8)));

#define N_IDX   8192
#define NW      8191            // weight vector length (valid k: 0..8190)
#define RPB     16              // rows per block (one WMMA M-tile)
#define STRIP   1024            // j columns per block
#define NSTRIPS (N_IDX / STRIP) // 8
#define NTILES  (N_IDX / RPB)   // 512
#define BLK     128             // 4 waves of 32
#define WBUF    1040            // LDS weight window: spans wbase..wbase+1038

// wave32 butterfly step: x += swizzle_xor(x); PAT must be an immediate.
template <int PAT>
__device__ __forceinline__ float swz_add(float x)
{
    int y = __builtin_amdgcn_ds_swizzle(__float_as_int(x), PAT);
    return x + __int_as_float(y);
}

__global__ __launch_bounds__(BLK)
void CorrelationLoss_main(const int* __restrict__ idx,
                          const float* __restrict__ w,
                          const float* __restrict__ D,
                          float* __restrict__ partial)
{
    const int I0   = blockIdx.x * RPB;
    const int S0   = blockIdx.y * STRIP;
    const int blin = blockIdx.y * gridDim.x + blockIdx.x;
    const int tid  = threadIdx.x;

    __shared__ float s_w[WBUF];
    __shared__ int   s_idx[STRIP];
    __shared__ float s_part[BLK / 32];

    // Strip entirely at/above the diagonal -> contributes nothing.
    if (S0 > I0 + 14) {
        if (tid == 0) partial[blin] = 0.0f;
        return;
    }

    // Weight indices touched: k = I0-1-j0 + n - kk  in  [I0-S0-1024, I0-S0+14].
    // Out-of-range k reads 0 from the pre-zeroed LDS window -> no conditionals
    // in the inner loop.
    const int wbase = I0 - S0 - 1024;
    const int k_lo  = max(0, wbase);
    const int k_hi  = min(NW - 1, I0 - S0 + 14);
    const int wlen  = k_hi - k_lo + 1;

    // Zero the whole window (covers the masked region), stage idx strip.
    for (int t = tid; t < WBUF; t += BLK) s_w[t] = 0.0f;
    for (int t = tid; t < STRIP; t += BLK) s_idx[t] = idx[S0 + t];
    __syncthreads();   // zero-fill visible before TDM overwrites valid part

    // Stage the valid w subrange into LDS with the Tensor Data Mover.
#if __has_builtin(__builtin_amdgcn_tensor_load_to_lds)
    if (tid < 32) {
        unsigned long long ga = (unsigned long long)(const void*)(w + k_lo);
        unsigned lds_off = (unsigned)(unsigned long long)(const void*)&s_w[k_lo - wbase];
        u32x4 g0;
        g0[0] = 1u;                                   // count=1, user descriptor
        g0[1] = lds_off;                              // lds_addr (bytes)
        g0[2] = (unsigned)ga;                         // global_addr[31:0]
        g0[3] = (unsigned)((ga >> 32) & 0x1FFFFFFull) // global_addr[56:32]
              | (2u << 30);                           // type = 2 ("image")
        i32x8 g1;
        g1[0] = (int)(2u << 16);                      // data_size = 2 -> 4 bytes
        g1[1] = (int)(((unsigned)wlen & 0xFFFFu) << 16);          // tensor_dim0 lo16
        g1[2] = (int)((((unsigned)wlen >> 16) & 0xFFFFu)          // tensor_dim0 hi16
              | (1u << 16));                                      // tensor_dim1 = 1
        g1[3] = (int)(((unsigned)wlen & 0xFFFFu) << 16);          // tile_dim0 = wlen
        g1[4] = 0;                                                // tile_dim1/2 unused
        g1[5] = wlen;                                             // tensor_dim0_stride
        g1[6] = 0;
        g1[7] = 0;
        i32x4 z4 = {0, 0, 0, 0};
#if __clang_major__ >= 23
        i32x8 z8 = {0, 0, 0, 0, 0, 0, 0, 0};
        __builtin_amdgcn_tensor_load_to_lds(g0, g1, z4, z4, z8, 0);
#else
        __builtin_amdgcn_tensor_load_to_lds(g0, g1, z4, z4, 0);
#endif
        __builtin_amdgcn_s_wait_tensorcnt(0);
    }
#else
    for (int t = tid; t < wlen; t += BLK) s_w[(k_lo - wbase) + t] = w[k_lo + t];
#endif
    __syncthreads();

    const int lane = tid & 31;
    const int arow = lane & 15;           // A-matrix row M held by this lane
    const int koff = (lane >> 4) << 1;    // K pair: lanes 0-15 -> K0/K1, 16-31 -> K2/K3

    // Force the wave index into an SGPR so loop control is scalar
    // (s_cmp/s_cbranch) and the unroll pragma can fire.
    const int wvu = __builtin_amdgcn_readfirstlane(tid >> 5);

    // 32-bit gather offsets from a uniform base: row*8192 + col < 2^27.
    const unsigned rbase = (unsigned)idx[I0 + arow] << 13;

    // Per-lane constant for the weight window index: kb = kconst - j0.
    const int kconst = I0 + arow - 1 - koff - wbase;

    v8f c = {0.f, 0.f, 0.f, 0.f, 0.f, 0.f, 0.f, 0.f};

    // Waves interleave 4-column chunks; scalar chunk count (diag-trimmed).
    const int jend    = min(S0 + STRIP, I0 + 15);
    const int j0start = S0 + 4 * wvu;
    const int nch     = (jend > j0start) ? ((jend - j0start + 15) >> 4) : 0;

    int j0 = j0start;
    #pragma unroll 8
    for (int t = 0; t < nch; ++t, j0 += 16) {
        // Logical columns j0+koff (c0) and j0+koff+1 (c1).  The mapping of
        // logical columns to physical K slots is arbitrary as long as A and B
        // agree, so put c1 in the LOW slot: then b.x sits at the lower LDS
        // address and ds_load_2addr lands directly in operand order (no swap).
        const int c0 = s_idx[(j0 - S0) + koff];
        const int c1 = s_idx[(j0 - S0) + koff + 1];
        v2f a;
        a.x = D[rbase + (unsigned)c1];
        a.y = D[rbase + (unsigned)c0];
        const int kb = kconst - j0;       // weight index for column j0+koff
        v2f b;
        b.x = s_w[kb - 1];                // weight for column j0+koff+1
        b.y = s_w[kb];
        // C[a,n] += sum_k G[a,k] * w[(I0+n)-1-(j0+k)]  -> diagonal = per-row sums
        c = __builtin_amdgcn_wmma_f32_16x16x4_f32(false, a, false, b,
                                                  (short)0, c, false, false);
    }

    // Extract diag(C): (m,m) for m<8 lives in lane m, elem m;
    // for m>=8 in lane m+16, elem m-8.
    float dval = 0.0f;
    #pragma unroll
    for (int e = 0; e < 8; ++e)
        if (lane == e || lane == e + 24) dval = c[e];

    // Wave32 butterfly reduction via ds_swizzle xor patterns.
    dval = swz_add<0x401F>(dval);   // xor 16
    dval = swz_add<0x201F>(dval);   // xor 8
    dval = swz_add<0x101F>(dval);   // xor 4
    dval = swz_add<0x081F>(dval);   // xor 2
    dval = swz_add<0x041F>(dval);   // xor 1

    if (lane == 0) s_part[tid >> 5] = dval;
    __syncthreads();
    if (tid == 0) {
        float s = 0.0f;
        #pragma unroll
        for (int v = 0; v < BLK / 32; ++v) s += s_part[v];
        partial[blin] = s;
    }
}

__global__ __launch_bounds__(256)
void CorrelationLoss_final(const float* __restrict__ partial, int nparts,
                           const float* __restrict__ w,
                           float* __restrict__ out)
{
    __shared__ float s0[256];
    __shared__ float s1[256];
    const int tid = threadIdx.x;

    float a = 0.0f;
    for (int i = tid; i < nparts; i += 256) a += partial[i];

    float b = 0.0f;
    for (int k = tid; k < NW - 1; k += 256) {
        float d = w[k] - w[k + 1];
        d = fminf(d, 0.0f) * 5.0f;
        b -= d;                         // weight_loss = -sum(d)
    }

    s0[tid] = a;
    s1[tid] = b;
    __syncthreads();
    for (int off = 128; off > 0; off >>= 1) {
        if (tid < off) { s0[tid] += s0[tid + off]; s1[tid] += s1[tid + off]; }
        __syncthreads();
    }
    if (tid == 0) {
        out[0] = -s0[0] / (float)(N_IDX - 1);   // correlation_loss
        out[1] = s1[0];                          // weight_loss
    }
}

extern "C" void kernel_launch(void* const* d_in, const int* in_sizes, int n_in,
                              void* d_out, int out_size, void* d_ws, size_t ws_size,
                              hipStream_t stream)
{
    (void)in_sizes; (void)n_in; (void)out_size; (void)ws_size;
    const int*   idx = (const int*)d_in[0];
    const float* w   = (const float*)d_in[1];
    const float* D   = (const float*)d_in[2];
    float*       out = (float*)d_out;
    float*       part = (float*)d_ws;   // NTILES*NSTRIPS floats = 16 KB scratch

    dim3 grid(NTILES, NSTRIPS);
    CorrelationLoss_main<<<grid, BLK, 0, stream>>>(idx, w, D, part);
    CorrelationLoss_final<<<1, 256, 0, stream>>>(part, NTILES * NSTRIPS, w, out);
}